// FilterShortRange_56891136802874
// MI455X (gfx1250) — compile-verified
//
#include <hip/hip_runtime.h>
#include <hip/hip_bf16.h>

// FilterShortRange: stable stream compaction of pairs with ||Rij|| <= 1.6.
// Bandwidth-bound (~1.44 GB traffic -> ~62us at 23.3 TB/s on MI455X).
// Wave32 ballot/mbcnt stable partition; deterministic 3-pass
// (count / scan / scatter), no atomics. Full blocks are compiled without
// bounds checks (straight-line, clause-packable loads); one guarded tail
// block handles the remainder.

#define CUTOFF2 2.56f  // 1.6^2 (compare r^2 vs cutoff^2; avoids sqrt)

constexpr int BLOCK = 256;            // 8 wave32s per workgroup
constexpr int TILES = 4;              // 4 x 256 = 1024 pairs per block
constexpr int PPB = BLOCK * TILES;    // pairs per block

// ---------------------------------------------------------------- pass 1 ----
// Per-block count of valid pairs.
template <bool FULL>
__global__ void fsr_count_kernel(const float* __restrict__ Rij,
                                 unsigned int* __restrict__ blockCounts,
                                 int P, int blockOff) {
  __shared__ unsigned int s_wave[BLOCK / 32];
  const int b = (int)blockIdx.x + blockOff;
  const int base = b * PPB;
  const int lane = threadIdx.x & 31;
  const int wave = threadIdx.x >> 5;

  unsigned int waveCnt = 0;  // uniform across the wave
#pragma unroll
  for (int k = 0; k < TILES; ++k) {
    const int pair = base + k * BLOCK + (int)threadIdx.x;
    bool valid = false;
    if (FULL || pair < P) {
      const float* r = Rij + 3 * (size_t)pair;  // global_load_b96, coalesced
      const float x = r[0], y = r[1], z = r[2];
      valid = (x * x + y * y + z * z) <= CUTOFF2;
    }
    const unsigned int bal = __builtin_amdgcn_ballot_w32(valid);  // wave32
    waveCnt += (unsigned int)__popc(bal);
  }
  if (lane == 0) s_wave[wave] = waveCnt;
  __syncthreads();
  if (threadIdx.x == 0) {
    unsigned int total = 0;
#pragma unroll
    for (int w = 0; w < BLOCK / 32; ++w) total += s_wave[w];
    blockCounts[b] = total;
  }
}

// ---------------------------------------------------------------- pass 2 ----
// Single-block exclusive scan of block counts; writes grand total (n_valid).
__global__ void fsr_scan_kernel(const unsigned int* __restrict__ counts,
                                unsigned int* __restrict__ offsets,
                                unsigned int* __restrict__ totalOut,
                                int nBlocks) {
  __shared__ unsigned int s_sums[BLOCK];
  const int t = threadIdx.x;
  const int per = (nBlocks + BLOCK - 1) / BLOCK;
  const int begin = t * per;
  const int end = min(begin + per, nBlocks);

  unsigned int segSum = 0;
  for (int i = begin; i < end; ++i) segSum += counts[i];
  s_sums[t] = segSum;
  __syncthreads();

  // Hillis-Steele inclusive scan over 256 segment sums (ds ops).
  for (int off = 1; off < BLOCK; off <<= 1) {
    unsigned int add = 0;
    if (t >= off) add = s_sums[t - off];
    __syncthreads();
    s_sums[t] += add;
    __syncthreads();
  }
  const unsigned int incl = s_sums[t];
  const unsigned int excl = incl - segSum;
  if (t == BLOCK - 1) totalOut[0] = incl;  // n_valid

  unsigned int running = excl;
  for (int i = begin; i < end; ++i) {
    offsets[i] = running;
    running += counts[i];
  }
}

// ---------------------------------------------------------------- pass 3 ----
// Scatter valid pairs to the front (stable), pad tail, emit pass-through
// copies and n_valid. Every output element is written exactly once per call.
template <bool FULL>
__global__ void fsr_scatter_kernel(const float* __restrict__ Rij,
                                   const int* __restrict__ idx_i,
                                   const int* __restrict__ idx_j,
                                   const unsigned int* __restrict__ blockOffsets,
                                   const unsigned int* __restrict__ totalPtr,
                                   float* __restrict__ out,
                                   int P, int blockOff) {
  __shared__ unsigned int s_wavecnt[BLOCK / 32];

  const size_t Ps = (size_t)P;
  float* outR = out;             // Rij_sr      [3P]
  float* outI = out + 3 * Ps;    // idx_i_sr    [P]
  float* outJ = out + 4 * Ps;    // idx_j_sr    [P]
  float* lrR  = out + 5 * Ps;    // Rij copy    [3P]
  float* lrI  = out + 8 * Ps;    // idx_i copy  [P]
  float* lrJ  = out + 9 * Ps;    // idx_j copy  [P]

  const int b = (int)blockIdx.x + blockOff;
  const int base = b * PPB;
  const int wave = threadIdx.x >> 5;

  const unsigned int nvalid = *totalPtr;                  // uniform -> s_load
  unsigned int vrun = blockOffsets[b];                    // valid dest cursor
  unsigned int irun = (unsigned int)base - vrun + nvalid; // invalid dest cursor

  if (b == 0 && threadIdx.x == 0)
    out[10 * Ps] = (float)nvalid;                         // n_valid output

#pragma unroll
  for (int k = 0; k < TILES; ++k) {
    const int tileStart = base + k * BLOCK;
    const int pair = tileStart + (int)threadIdx.x;
    const bool inb = FULL || (pair < P);

    float x = 0.f, y = 0.f, z = 0.f;
    int ii = -1, jj = -1;
    if (inb) {
      const float* r = Rij + 3 * (size_t)pair;            // global_load_b96
      x = r[0]; y = r[1]; z = r[2];
      ii = idx_i[pair];
      jj = idx_j[pair];
      // Pass-through copies (streaming: non-temporal to spare L2).
      float* lr = lrR + 3 * (size_t)pair;
      __builtin_nontemporal_store(x, lr + 0);
      __builtin_nontemporal_store(y, lr + 1);
      __builtin_nontemporal_store(z, lr + 2);
      __builtin_nontemporal_store((float)ii, lrI + pair);
      __builtin_nontemporal_store((float)jj, lrJ + pair);
    }
    const bool valid = inb && (x * x + y * y + z * z) <= CUTOFF2;

    // Stable intra-tile rank: wave32 ballot + mbcnt + cross-wave LDS totals.
    const unsigned int bal = __builtin_amdgcn_ballot_w32(valid);
    const unsigned int rankInWave = __builtin_amdgcn_mbcnt_lo(bal, 0u);
    const unsigned int waveTotal = (unsigned int)__popc(bal);
    if ((threadIdx.x & 31) == 0) s_wavecnt[wave] = waveTotal;
    __syncthreads();
    unsigned int waveBase = 0, tileTotal = 0;
#pragma unroll
    for (int w = 0; w < BLOCK / 32; ++w) {
      const unsigned int c = s_wavecnt[w];
      if (w < wave) waveBase += c;
      tileTotal += c;
    }

    if (inb) {
      const unsigned int validBefore = waveBase + rankInWave;
      if (valid) {
        const size_t dst = (size_t)(vrun + validBefore);
        float* o = outR + 3 * dst;
        __builtin_nontemporal_store(x, o + 0);
        __builtin_nontemporal_store(y, o + 1);
        __builtin_nontemporal_store(z, o + 2);
        __builtin_nontemporal_store((float)ii, outI + dst);
        __builtin_nontemporal_store((float)jj, outJ + dst);
      } else {
        // Earlier lanes in the tile are in-bounds whenever this lane is.
        const unsigned int invBefore = (unsigned int)threadIdx.x - validBefore;
        const size_t dst = (size_t)(irun + invBefore);
        float* o = outR + 3 * dst;
        __builtin_nontemporal_store(0.0f, o + 0);
        __builtin_nontemporal_store(0.0f, o + 1);
        __builtin_nontemporal_store(0.0f, o + 2);
        __builtin_nontemporal_store(-1.0f, outI + dst);
        __builtin_nontemporal_store(-1.0f, outJ + dst);
      }
    }
    __syncthreads();  // s_wavecnt reused next tile

    unsigned int tileInb;
    if (FULL) {
      tileInb = (unsigned int)BLOCK;
    } else {
      const int rem = P - tileStart;
      tileInb = (rem >= BLOCK) ? (unsigned int)BLOCK
                               : (unsigned int)(rem > 0 ? rem : 0);
    }
    vrun += tileTotal;
    irun += tileInb - tileTotal;
  }
}

// ------------------------------------------------------------------ launch --
extern "C" void kernel_launch(void* const* d_in, const int* in_sizes, int n_in,
                              void* d_out, int out_size, void* d_ws, size_t ws_size,
                              hipStream_t stream) {
  const float* Rij   = (const float*)d_in[0];
  const int*   idx_i = (const int*)d_in[1];
  const int*   idx_j = (const int*)d_in[2];
  float* out = (float*)d_out;

  const int P = in_sizes[1];  // N_PAIRS (idx_i flat count)
  const int fullBlocks = P / PPB;
  const bool hasTail = (P % PPB) != 0;
  const int nBlocks = fullBlocks + (hasTail ? 1 : 0);

  unsigned int* counts  = (unsigned int*)d_ws;
  unsigned int* offsets = counts + nBlocks;
  unsigned int* total   = offsets + nBlocks;

  // Pass 1: per-block valid counts (full blocks unguarded + guarded tail).
  if (fullBlocks > 0)
    fsr_count_kernel<true><<<fullBlocks, BLOCK, 0, stream>>>(Rij, counts, P, 0);
  if (hasTail)
    fsr_count_kernel<false><<<1, BLOCK, 0, stream>>>(Rij, counts, P, fullBlocks);

  // Pass 2: exclusive scan of block counts + n_valid.
  fsr_scan_kernel<<<1, BLOCK, 0, stream>>>(counts, offsets, total, nBlocks);

  // Pass 3: stable scatter + tail padding + pass-through copies + n_valid.
  if (fullBlocks > 0)
    fsr_scatter_kernel<true><<<fullBlocks, BLOCK, 0, stream>>>(
        Rij, idx_i, idx_j, offsets, total, out, P, 0);
  if (hasTail)
    fsr_scatter_kernel<false><<<1, BLOCK, 0, stream>>>(
        Rij, idx_i, idx_j, offsets, total, out, P, fullBlocks);
}